// SubnetGate_89678917140888
// MI455X (gfx1250) — compile-verified
//
#include <hip/hip_runtime.h>

typedef __attribute__((ext_vector_type(2))) float v2f;
typedef __attribute__((ext_vector_type(8))) float v8f;

#define D_IN   512
#define D_OUT  512
#define NSUB   8
#define TILE_M 128
#define TILE_N 64
#define TILE_K 32

// ---------------- binning kernels ----------------

__global__ void sg_init(int* __restrict__ counts) {
    if (threadIdx.x < NSUB) counts[threadIdx.x] = 0;
}

__global__ void sg_hist(const int* __restrict__ groups, int* __restrict__ counts, int nrows) {
    int b = blockIdx.x * blockDim.x + threadIdx.x;
    if (b < nrows) {
        int e = groups[b * 2];          // group_col = 0, G = 2
        atomicAdd(&counts[e], 1);
    }
}

__global__ void sg_scan(const int* __restrict__ counts, int* __restrict__ offsets,
                        int* __restrict__ cursors) {
    if (threadIdx.x == 0 && blockIdx.x == 0) {
        int s = 0;
        for (int e = 0; e < NSUB; ++e) {
            offsets[e] = s;
            cursors[e] = s;
            s += counts[e];
        }
        offsets[NSUB] = s;
    }
}

__global__ void sg_scatter(const int* __restrict__ groups, int* __restrict__ cursors,
                           int* __restrict__ perm, int nrows) {
    int b = blockIdx.x * blockDim.x + threadIdx.x;
    if (b < nrows) {
        int e = groups[b * 2];
        int p = atomicAdd(&cursors[e], 1);
        perm[p] = b;
    }
}

// ---------------- grouped GEMM via V_WMMA_F32_16X16X4_F32 ----------------
// Block: 256 threads = 8 wave32s. Tile: M=128 (wave w owns rows w*16..w*16+15),
// N=64 (4 WMMA n-subtiles of 16), K staged 32 at a time through LDS.
// Software-pipelined: global loads for tile k+1 are issued before the WMMA
// block for tile k, and only waited at the next LDS store.

__global__ __launch_bounds__(256) void sg_gemm(
    const float* __restrict__ x, const float* __restrict__ W,
    const float* __restrict__ bias, const int* __restrict__ offsets,
    const int* __restrict__ perm, float* __restrict__ out)
{
    __shared__ __align__(16) float As[TILE_M * TILE_K];  // 16 KB
    __shared__ __align__(16) float Bs[TILE_K * TILE_N];  //  8 KB

    // Map linear tile id -> (expert, row window) using bin offsets (uniform per block).
    int t = blockIdx.x;
    int e = -1, rs = 0, re = 0;
    int acc = 0;
    for (int i = 0; i < NSUB; ++i) {
        int lo = offsets[i], hi = offsets[i + 1];
        int tiles = (hi - lo + TILE_M - 1) / TILE_M;
        if (t < acc + tiles) { e = i; rs = lo + (t - acc) * TILE_M; re = hi; break; }
        acc += tiles;
    }
    if (e < 0) return;                       // uniform early-exit for surplus blocks

    const int tid   = threadIdx.x;
    const int wave  = tid >> 5;
    const int lane  = tid & 31;
    const int nbase = blockIdx.y * TILE_N;
    const float* We = W + (size_t)e * D_IN * D_OUT;

    // A staging: thread covers row (tid>>1), 16 floats starting at (tid&1)*16.
    const int arow = tid >> 1;
    const int acol = (tid & 1) * 16;
    const int aslot = rs + arow;
    const int grow = (aslot < re) ? perm[aslot] : -1;
    const float* ap = (grow >= 0) ? (x + (size_t)grow * D_IN + acol) : x;

    // B staging: thread covers k-row (tid>>3), 8 floats starting at (tid&7)*8.
    const int bk = tid >> 3;
    const int bc = (tid & 7) * 8;
    const float* bp = We + (size_t)bk * D_OUT + nbase + bc;

    v8f c0 = {}, c1 = {}, c2 = {}, c3 = {};

    const int m    = lane & 15;
    const int half = lane >> 4;              // lanes 16-31 hold K+2 (A) / rows +2 (B)

    // ---- pipeline prologue: stage tile kt=0 into registers ----
    float4 a0, a1, a2, a3, b0, b1;
    a0 = a1 = a2 = a3 = make_float4(0.f, 0.f, 0.f, 0.f);
    if (grow >= 0) {
        const float4* s = (const float4*)ap;
        a0 = s[0]; a1 = s[1]; a2 = s[2]; a3 = s[3];
    }
    {
        const float4* bsrc = (const float4*)bp;
        b0 = bsrc[0];
        b1 = bsrc[1];
    }

    for (int kt = 0; kt < D_IN; kt += TILE_K) {
        // ---- staged registers -> LDS ----
        __syncthreads();                     // previous tile's WMMAs done reading LDS
        {
            float4* ad = (float4*)&As[arow * TILE_K + acol];
            ad[0] = a0; ad[1] = a1; ad[2] = a2; ad[3] = a3;
            float4* bd = (float4*)&Bs[bk * TILE_N + bc];
            bd[0] = b0; bd[1] = b1;
        }
        __syncthreads();

        // ---- issue global loads for next tile (waited at next LDS store) ----
        const int ktn = kt + TILE_K;
        if (ktn < D_IN) {
            if (grow >= 0) {
                const float4* s = (const float4*)(ap + ktn);
                a0 = s[0]; a1 = s[1]; a2 = s[2]; a3 = s[3];
            }
            const float4* bsrc = (const float4*)(bp + (size_t)ktn * D_OUT);
            b0 = bsrc[0];
            b1 = bsrc[1];
        }

        // ---- WMMA over the K-tile: 8 k-steps x 4 n-subtiles ----
        const float* arowp = &As[(wave * 16 + m) * TILE_K];
        #pragma unroll
        for (int ks = 0; ks < TILE_K / 4; ++ks) {
            // A 16x4 fragment: lane (m, half) holds K = half*2 + {0,1}
            v2f a;
            a.x = arowp[ks * 4 + half * 2 + 0];
            a.y = arowp[ks * 4 + half * 2 + 1];

            const float* brow0 = &Bs[(ks * 4 + half * 2) * TILE_N + m];
            const float* brow1 = brow0 + TILE_N;
            v2f b;

            b.x = brow0[0];  b.y = brow1[0];
            c0 = __builtin_amdgcn_wmma_f32_16x16x4_f32(false, a, false, b, (short)0, c0, false, false);
            b.x = brow0[16]; b.y = brow1[16];
            c1 = __builtin_amdgcn_wmma_f32_16x16x4_f32(false, a, false, b, (short)0, c1, false, false);
            b.x = brow0[32]; b.y = brow1[32];
            c2 = __builtin_amdgcn_wmma_f32_16x16x4_f32(false, a, false, b, (short)0, c2, false, false);
            b.x = brow0[48]; b.y = brow1[48];
            c3 = __builtin_amdgcn_wmma_f32_16x16x4_f32(false, a, false, b, (short)0, c3, false, false);
        }
    }

    // ---- epilogue: bias + scatter back to original row order ----
    const float bias0 = bias[(size_t)e * D_OUT + nbase + 0  + m];
    const float bias1 = bias[(size_t)e * D_OUT + nbase + 16 + m];
    const float bias2 = bias[(size_t)e * D_OUT + nbase + 32 + m];
    const float bias3 = bias[(size_t)e * D_OUT + nbase + 48 + m];

    #pragma unroll
    for (int v = 0; v < 8; ++v) {
        int mloc = v + half * 8;             // C layout: VGPR v -> M=v (lo half) / v+8 (hi half)
        int slot = rs + wave * 16 + mloc;
        if (slot < re) {
            int g = perm[slot];
            float* o = out + (size_t)g * D_OUT + nbase;
            o[0  + m] = c0[v] + bias0;
            o[16 + m] = c1[v] + bias1;
            o[32 + m] = c2[v] + bias2;
            o[48 + m] = c3[v] + bias3;
        }
    }
}

// ---------------- launch ----------------

extern "C" void kernel_launch(void* const* d_in, const int* in_sizes, int n_in,
                              void* d_out, int out_size, void* d_ws, size_t ws_size,
                              hipStream_t stream) {
    const float* x      = (const float*)d_in[0];   // (B, 512) f32
    const int*   groups = (const int*)d_in[1];     // (B, 2) i32
    const float* W      = (const float*)d_in[2];   // (8, 512, 512) f32
    const float* bias   = (const float*)d_in[3];   // (8, 512) f32
    float*       out    = (float*)d_out;           // (B, 512) f32

    const int nrows = in_sizes[1] / 2;             // B = 16384

    int* wsI     = (int*)d_ws;
    int* counts  = wsI;            // 8
    int* offsets = wsI + 8;        // 9
    int* cursors = wsI + 17;       // 8
    int* perm    = wsI + 32;       // nrows

    sg_init<<<1, 32, 0, stream>>>(counts);
    sg_hist<<<(nrows + 255) / 256, 256, 0, stream>>>(groups, counts, nrows);
    sg_scan<<<1, 1, 0, stream>>>(counts, offsets, cursors);
    sg_scatter<<<(nrows + 255) / 256, 256, 0, stream>>>(groups, cursors, perm, nrows);

    const int mtiles = (nrows + TILE_M - 1) / TILE_M + NSUB;   // upper bound incl. partial tiles
    dim3 grid(mtiles, D_OUT / TILE_N);
    sg_gemm<<<grid, 256, 0, stream>>>(x, W, bias, offsets, perm, out);
}